// FFTConv2d_88450556494620
// MI455X (gfx1250) — compile-verified
//
#include <hip/hip_runtime.h>
#include <hip/hip_bf16.h>

// FFT conv2d == circular conv of reflect-padded 256x256 input with 5x5 kernel.
// Implemented as implicit GEMM per kernel position using fp32 WMMA
// (v_wmma_f32_16x16x4_f32) to match the fp32 reference precision exactly.
//
// LDS layouts interleave ci-pairs so every WMMA A/B fragment is one aligned
// ds_load_b64 into an even VGPR pair (no v_mov glue, conflict-free banks).
// Strip fill uses the CDNA5 async global->LDS path (ASYNCcnt).

typedef float v2f __attribute__((ext_vector_type(2)));
typedef float v8f __attribute__((ext_vector_type(8)));

typedef __attribute__((address_space(1))) int as1_int;   // global
typedef __attribute__((address_space(3))) int as3_int;   // LDS

#define HIMG   252            // image height/width
#define XST2   544            // strip row stride in floats (256 cols * 2 interleave,
                              // padded so 5*XST2 % 64 == 32 -> opposite bank half)
#define PCIST  (5 * XST2)     // per ci-pair strip size = 2720 floats (2720%64==32)
#define ASTP   96             // weight-slice ci-pair stride (96%64==32)
#define STRIPF (16 * PCIST)   // 43520 floats
#define LDSF   (STRIPF + 16 * ASTP)   // 45056 floats = 180,224 B (< 320KB/WGP)

#if __has_builtin(__builtin_amdgcn_global_load_async_to_lds_b32) && \
    __has_builtin(__builtin_amdgcn_s_wait_asynccnt)
#define USE_ASYNC_LDS 1
#else
#define USE_ASYNC_LDS 0
#endif

__device__ __forceinline__ int reflect_idx(int s) {
    s = (s < 0) ? -s : s;            // xp[0]=x[2], xp[1]=x[1]
    s = (s > 251) ? 502 - s : s;     // xp[254]=x[250], xp[255]=x[249]
    return s;
}

// Pre-transpose weights: w[co][ci][ky][kx] -> wt[kpos][ci][co]  (kpos = ky*5+kx)
__global__ __launch_bounds__(256) void FFTConv2d_wt_kernel(
        const float* __restrict__ w, float* __restrict__ wt) {
    int i = blockIdx.x * 256 + threadIdx.x;      // 0 .. 25599
    if (i >= 32 * 32 * 25) return;
    int co  = i / (32 * 25);
    int rem = i % (32 * 25);
    int ci  = rem / 25;
    int kp  = rem % 25;
    wt[(kp * 32 + ci) * 32 + co] = w[i];
}

__global__ __launch_bounds__(256) void FFTConv2d_wmma_kernel(
        const float* __restrict__ x,   // [8,32,252,252]
        const float* __restrict__ wt,  // [25,32,32] (ci-major, co contiguous)
        float* __restrict__ out) {     // [8,32,252,252]
    extern __shared__ float lds[];
    float* sX = lds;            // [16 ci-pair][5 ry][256 col][2] input strip
    float* sA = lds + STRIPF;   // [16 ci-pair][32 co][2] weight slice

    const int b   = blockIdx.x / HIMG;
    const int y   = blockIdx.x % HIMG;
    const int tid = threadIdx.x;

    // ---- strip fill: rows (y+1 .. y+5) mod 256, 32 ci, 256 padded cols.
    // 32*5*256 = 40960 elements, 160 per thread, coalesced along columns.
    for (int e = 0; e < 160; ++e) {
        int flat = tid + e * 256;
        int ci   = flat / 1280;          // 1280 = 5*256
        int rem  = flat - ci * 1280;
        int ry   = rem >> 8;             // 0..4
        int j    = rem & 255;            // padded column
        int rr   = (y + 1 + ry) & 255;   // circular wrap of the FFT grid
        int sr   = reflect_idx(rr - 2);  // reflect pad -> source row
        int sc   = reflect_idx(j - 2);   // reflect pad -> source col
        int src  = ((b * 32 + ci) * HIMG + sr) * HIMG + sc;
        int dst  = (ci >> 1) * PCIST + ry * XST2 + j * 2 + (ci & 1);
#if USE_ASYNC_LDS
        __builtin_amdgcn_global_load_async_to_lds_b32(
            (as1_int*)(int*)(x + src),   // global src (addrspace(1) int*)
            (as3_int*)(sX + dst),        // LDS dst (addrspace(3) int*)
            /*offset=*/0, /*cpol=*/0);
#else
        sX[dst] = x[src];
#endif
    }
#if USE_ASYNC_LDS
    __builtin_amdgcn_s_wait_asynccnt(0);   // LDS writes landed before barrier
#endif

    const int lane = tid & 31;
    const int l16  = lane & 15;
    const int hi   = lane >> 4;          // lanes 16-31 hold K+2 / M+8 per ISA layout
    const int warp = tid >> 5;           // 8 waves; wave owns x-tiles {32w, 32w+16}

    const v8f vzero = {0.f, 0.f, 0.f, 0.f, 0.f, 0.f, 0.f, 0.f};
    v8f acc[2][2];                       // [x-tile][co-half]
    acc[0][0] = vzero; acc[0][1] = vzero;
    acc[1][0] = vzero; acc[1][1] = vzero;

    for (int kp = 0; kp < 25; ++kp) {
        __syncthreads();                 // previous slice fully consumed
        // ---- load 32x32 weight slice (coalesced, 4 elems/thread) into sA
        {
            const float* ws = wt + kp * 1024;
#pragma unroll
            for (int e = 0; e < 4; ++e) {
                int f  = tid + e * 256;
                int ci = f >> 5;
                int co = f & 31;
                sA[(ci >> 1) * ASTP + co * 2 + (ci & 1)] = ws[f];
            }
            if (kp + 1 < 25)             // global_prefetch_b8 of next slice
                __builtin_prefetch(wt + (kp + 1) * 1024 + tid * 4, 0, 1);
        }
        __syncthreads();                 // strip (1st iter) + slice visible

        const int ky = kp / 5;
        const int kx = kp - ky * 5;
        const int ry = 4 - ky;           // strip row holding image row y+5-ky
        const int colbase = 5 - kx;      // input col = xo + 5 - kx (mod 256)

#pragma unroll
        for (int kk = 0; kk < 8; ++kk) { // K = 32 ci in 8 steps of 4
            const int p = 2 * kk + hi;   // ci-pair index: (4kk + 2hi)/2
            const float* sa = sA + p * ASTP;
            const v2f a0 = *(const v2f*)(sa + l16 * 2);         // co 0-15
            const v2f a1 = *(const v2f*)(sa + (l16 + 16) * 2);  // co 16-31
            const float* sb = sX + p * PCIST + ry * XST2;
#pragma unroll
            for (int t = 0; t < 2; ++t) {
                const int xo0 = warp * 32 + t * 16;
                const int col = (xo0 + l16 + colbase) & 255;
                const v2f bv = *(const v2f*)(sb + col * 2);
                acc[t][0] = __builtin_amdgcn_wmma_f32_16x16x4_f32(
                    false, a0, false, bv, (short)0, acc[t][0], false, false);
                acc[t][1] = __builtin_amdgcn_wmma_f32_16x16x4_f32(
                    false, a1, false, bv, (short)0, acc[t][1], false, false);
            }
        }
    }

    // ---- store D: VGPR r holds M = r (lanes 0-15) / r+8 (lanes 16-31), N = l16
    for (int t = 0; t < 2; ++t) {
        const int xo = warp * 32 + t * 16 + l16;
        if (xo < HIMG) {
            for (int h = 0; h < 2; ++h) {
#pragma unroll
                for (int r = 0; r < 8; ++r) {
                    const int co = h * 16 + r + 8 * hi;
                    out[((b * 32 + co) * HIMG + y) * HIMG + xo] = acc[t][h][r];
                }
            }
        }
    }
}

extern "C" void kernel_launch(void* const* d_in, const int* in_sizes, int n_in,
                              void* d_out, int out_size, void* d_ws, size_t ws_size,
                              hipStream_t stream) {
    const float* x  = (const float*)d_in[0];   // [8,32,252,252] fp32
    const float* w  = (const float*)d_in[1];   // [32,32,5,5] fp32
    float*       o  = (float*)d_out;           // [8,32,252,252] fp32
    float*       wt = (float*)d_ws;            // needs 25*32*32*4 = 102,400 B scratch

    // 1) transpose weights into [kpos][ci][co] for coalesced LDS fills
    FFTConv2d_wt_kernel<<<100, 256, 0, stream>>>(w, wt);

    // 2) implicit-GEMM conv: one workgroup per (batch, output row)
    const size_t ldsBytes = (size_t)LDSF * sizeof(float);   // 180,224 B / WGP
    FFTConv2d_wmma_kernel<<<8 * HIMG, 256, ldsBytes, stream>>>(x, wt, o);
}